// HausdorffLoss_75290776698925
// MI455X (gfx1250) — compile-verified
//
#include <hip/hip_runtime.h>
#include <stdint.h>

// Problem constants (from reference: B=8, C=4, H=W=512, 30 DT iterations)
#define HH   512
#define WW   512
#define NCH  4
#define NB   8
#define HW   (HH * WW)
#define ITERS 30
#define FARD 200.0f
#define BIGD 1.0e30f      // "+inf" padding outside the image, matches -inf maxpool pad

// Tiling: 64x64 core + 30-halo => all 30 Jacobi iterations run in LDS,
// core region is exact (influence radius == iterations <= halo).
#define TILE 64
#define HALO 30
#define PT   (TILE + 2 * HALO)   // 124
#define PTC  (PT * PT)           // 15376
#define NTHR 256
#define CORE_PER_THREAD ((TILE * TILE) / NTHR)  // 16

typedef __attribute__((ext_vector_type(2))) float v2f;
typedef __attribute__((ext_vector_type(8))) float v8f;
typedef __attribute__((ext_vector_type(4))) unsigned int u32x4;
typedef __attribute__((ext_vector_type(8))) int i32x8;
typedef __attribute__((ext_vector_type(4))) int i32x4;

#if __has_builtin(__builtin_amdgcn_tensor_load_to_lds)
#define USE_TDM 1
#else
#define USE_TDM 0
#endif

#if __has_builtin(__builtin_amdgcn_global_load_async_to_lds_b32)
#define USE_ASYNC_LDS 1
typedef __attribute__((address_space(1))) int* as1_ip;
typedef __attribute__((address_space(3))) int* as3_ip;
#else
#define USE_ASYNC_LDS 0
#endif

// Exact f32 sum of the 32 lane values using V_WMMA_F32_16X16X4_F32:
// A[m][k] holds the lane partials (K=0 from lanes 0-15, K=2 from lanes 16-31),
// B = all-ones => D[m][n] = acc_m + acc_{m+16}. Each lane then holds 8 row-sums;
// lanes n and n^16 hold disjoint row halves, so one shfl_xor finishes the sum.
__device__ __forceinline__ float wave_sum_wmma(float acc) {
    v2f a; a[0] = acc;  a[1] = 0.0f;
    v2f b; b[0] = 1.0f; b[1] = 1.0f;
    v8f c = {};
    v8f d = __builtin_amdgcn_wmma_f32_16x16x4_f32(false, a, false, b,
                                                  (short)0, c, false, false);
    float s = d[0] + d[1] + d[2] + d[3] + d[4] + d[5] + d[6] + d[7];
    s += __shfl_xor(s, 16, 32);
    return s;
}

__global__ __launch_bounds__(NTHR) void hausdorff_tile_kernel(
        const float* __restrict__ logits,
        const int*   __restrict__ y_true,
        float*       __restrict__ partials) {
    __shared__ float sA[PTC];           // distance field (ping)
    __shared__ float sB[PTC];           // row-min scratch (pong); also int staging
    __shared__ unsigned char sCls[PTC]; // 0=outside image, 1=background, 2=foreground
    __shared__ float sRed[NTHR / 32];

    const int tid = threadIdx.x;
    const int blk = blockIdx.x;
    const int b   = blk >> 6;          // 8 batches x 64 tiles
    const int t   = blk & 63;
    const int y0  = (t >> 3) * TILE;
    const int x0  = (t & 7) * TILE;
    const int py0 = y0 - HALO;
    const int px0 = x0 - HALO;
    const size_t ybase = (size_t)b * HW;
    const size_t lbase = (size_t)b * NCH * HW;

    // ---- Phase A: classify every padded-tile cell ------------------------
#if USE_TDM
    {
        // Clamp the halo rect to the image; the TDM loads the clamped 2D tile
        // (packed rows of width cw) from the 512-wide label image into LDS.
        const int oy = py0 > 0 ? py0 : 0;
        const int ox = px0 > 0 ? px0 : 0;
        const int ey = (py0 + PT) < HH ? (py0 + PT) : HH;
        const int ex = (px0 + PT) < WW ? (px0 + PT) : WW;
        const int cw = ex - ox;
        const int ch = ey - oy;
        int* sBint = (int*)sB;
        if (__builtin_amdgcn_readfirstlane(tid) == 0) {   // wave 0, scalar branch
            unsigned long long gaddr =
                (unsigned long long)(const void*)&y_true[ybase + (size_t)oy * WW + ox];
            unsigned ldsOff = (unsigned)(unsigned long long)(void*)sBint;
            u32x4 g0;
            g0[0] = 1u;                                    // count=1 (valid), user mode
            g0[1] = ldsOff;                                // lds_addr (bytes)
            g0[2] = (unsigned)(gaddr & 0xFFFFFFFFull);     // global_addr[31:0]
            g0[3] = (unsigned)((gaddr >> 32) & 0x1FFFFFFull) | (2u << 30); // [56:32] | type=2
            i32x8 g1;
            g1[0] = (int)(2u << 16);                       // wg_mask=0, data_size=2 (4B)
            g1[1] = (int)(((unsigned)cw & 0xFFFFu) << 16); // tensor_dim0[15:0] (bits 63:48)
            g1[2] = (int)(((unsigned)ch & 0xFFFFu) << 16); // dim0 hi=0 | tensor_dim1[15:0]
            g1[3] = (int)(((unsigned)cw & 0xFFFFu) << 16); // dim1 hi=0 | tile_dim0
            g1[4] = (int)((unsigned)ch & 0xFFFFu);         // tile_dim1 | tile_dim2=0
            g1[5] = WW;                                    // tensor_dim0_stride = 512
            g1[6] = 0;                                     // stride hi | dim1_stride lo
            g1[7] = 0;                                     // dim1_stride hi
            i32x4 gz = {0, 0, 0, 0};                       // 2D tensor: groups 2/3 unused
            i32x8 gz8 = {0, 0, 0, 0, 0, 0, 0, 0};
            __builtin_amdgcn_tensor_load_to_lds(g0, g1, gz, gz, gz8, 0);
            __builtin_amdgcn_s_wait_tensorcnt(0);
        }
        __syncthreads();
        for (int i = tid; i < PTC; i += NTHR) {
            int py = i / PT, px = i - py * PT;
            int gy = py0 + py, gx = px0 + px;
            unsigned char c = 0;
            if ((unsigned)gy < HH && (unsigned)gx < WW)
                c = (sBint[(gy - oy) * cw + (gx - ox)] > 0) ? (unsigned char)2
                                                            : (unsigned char)1;
            sCls[i] = c;
        }
    }
#elif USE_ASYNC_LDS
    {
        int* sBint = (int*)sB;
        for (int i = tid; i < PTC; i += NTHR) {
            int py = i / PT, px = i - py * PT;
            int gy = py0 + py, gx = px0 + px;
            if ((unsigned)gy < HH && (unsigned)gx < WW) {
                const int* gp = &y_true[ybase + (size_t)gy * WW + gx];
                int*       lp = &sBint[i];
                __builtin_amdgcn_global_load_async_to_lds_b32(
                    (as1_ip)(unsigned long long)gp,
                    (as3_ip)(unsigned)(unsigned long long)lp,
                    0, 0);
            }
        }
#if __has_builtin(__builtin_amdgcn_s_wait_asynccnt)
        __builtin_amdgcn_s_wait_asynccnt(0);
#else
        asm volatile("s_wait_asynccnt 0" ::: "memory");
#endif
        __syncthreads();
        for (int i = tid; i < PTC; i += NTHR) {
            int py = i / PT, px = i - py * PT;
            int gy = py0 + py, gx = px0 + px;
            unsigned char c = 0;
            if ((unsigned)gy < HH && (unsigned)gx < WW)
                c = (sBint[i] > 0) ? (unsigned char)2 : (unsigned char)1;
            sCls[i] = c;
        }
    }
#else
    for (int i = tid; i < PTC; i += NTHR) {
        int py = i / PT, px = i - py * PT;
        int gy = py0 + py, gx = px0 + px;
        unsigned char c = 0;
        if ((unsigned)gy < HH && (unsigned)gx < WW)
            c = (y_true[ybase + (size_t)gy * WW + gx] > 0) ? (unsigned char)2
                                                           : (unsigned char)1;
        sCls[i] = c;
    }
#endif
    __syncthreads();

    // Warm L2 with this tile's logits while the distance transforms run.
    {
        const float* p = &logits[lbase + (size_t)y0 * WW + x0 + tid];
        __builtin_prefetch(p, 0, 1);
        __builtin_prefetch(p + HW, 0, 1);
        __builtin_prefetch(p + 2 * HW, 0, 1);
        __builtin_prefetch(p + 3 * HW, 0, 1);
    }

    // ---- Phase B: two chamfer distance transforms, all in LDS ------------
    // Jacobi: d <- min(d, minpool3x3(d) + 1), fg pinned at 0 automatically
    // (monotone non-increasing update, fg starts at 0). Separable 3x3 min.
    float wsum[CORE_PER_THREAD];
    for (int pass = 0; pass < 2; ++pass) {
        const unsigned char fg = (pass == 0) ? (unsigned char)2 : (unsigned char)1;
        for (int i = tid; i < PTC; i += NTHR) {
            unsigned char c = sCls[i];
            sA[i] = (c == 0) ? BIGD : ((c == fg) ? 0.0f : FARD);
        }
        __syncthreads();
        for (int it = 0; it < ITERS; ++it) {
            for (int i = tid; i < PTC; i += NTHR) {     // horizontal min3
                int px = i % PT;
                float v = sA[i];
                if (px > 0)      v = fminf(v, sA[i - 1]);
                if (px < PT - 1) v = fminf(v, sA[i + 1]);
                sB[i] = v;
            }
            __syncthreads();
            for (int i = tid; i < PTC; i += NTHR) {     // vertical min3 + update
                float v = sB[i];
                if (i >= PT)       v = fminf(v, sB[i - PT]);
                if (i < PTC - PT)  v = fminf(v, sB[i + PT]);
                sA[i] = fminf(sA[i], v + 1.0f);
            }
            __syncthreads();
        }
#pragma unroll
        for (int k = 0; k < CORE_PER_THREAD; ++k) {     // harvest core d^2
            int j  = k * NTHR + tid;
            int cy = j >> 6, cx = j & 63;
            float dv = sA[(cy + HALO) * PT + (cx + HALO)];
            if (pass == 0) wsum[k] = dv * dv;
            else           wsum[k] += dv * dv;          // = gt_dist^2 + bg_dist^2
        }
        __syncthreads();
    }

    // ---- Phase C: softmax-max + weighted squared error + reduction -------
    float acc = 0.0f;
#pragma unroll
    for (int k = 0; k < CORE_PER_THREAD; ++k) {
        int j  = k * NTHR + tid;
        int cy = j >> 6, cx = j & 63;
        int gy = y0 + cy, gx = x0 + cx;
        size_t off = lbase + (size_t)gy * WW + gx;
        float l0 = logits[off];
        float l1 = logits[off + HW];
        float l2 = logits[off + 2 * HW];
        float l3 = logits[off + 3 * HW];
        float M  = fmaxf(fmaxf(l0, l1), fmaxf(l2, l3));
        float e0 = __expf(l0 - M), e1 = __expf(l1 - M);
        float e2 = __expf(l2 - M), e3 = __expf(l3 - M);
        float p  = fmaxf(fmaxf(e1, e2), e3) / (e0 + e1 + e2 + e3);
        float g  = (sCls[(cy + HALO) * PT + (cx + HALO)] == 2) ? 1.0f : 0.0f;
        float d  = p - g;
        acc += d * d * wsum[k];
    }

    // EXEC is all-1s here (uniform control flow) -> WMMA is legal.
    float ws = wave_sum_wmma(acc);
    const int wave = tid >> 5;
    const int lane = tid & 31;
    if (lane == 0) sRed[wave] = ws;
    __syncthreads();
    if (tid == 0) {
        float s = 0.0f;
#pragma unroll
        for (int w = 0; w < NTHR / 32; ++w) s += sRed[w];
        partials[blk] = s;
    }
}

__global__ __launch_bounds__(256) void final_reduce_kernel(
        const float* __restrict__ partials, float* __restrict__ out) {
    __shared__ float s[256];
    float a = 0.0f;
    for (int i = threadIdx.x; i < NB * 64; i += 256) a += partials[i];
    s[threadIdx.x] = a;
    __syncthreads();
    for (int w = 128; w > 0; w >>= 1) {
        if (threadIdx.x < (unsigned)w) s[threadIdx.x] += s[threadIdx.x + w];
        __syncthreads();
    }
    if (threadIdx.x == 0) out[0] = s[0] * (1.0f / (float)((size_t)NB * HW));
}

extern "C" void kernel_launch(void* const* d_in, const int* in_sizes, int n_in,
                              void* d_out, int out_size, void* d_ws, size_t ws_size,
                              hipStream_t stream) {
    const float* logits = (const float*)d_in[0];   // (8,4,512,512) f32
    const int*   y_true = (const int*)d_in[1];     // (8,1,512,512) i32
    float* out      = (float*)d_out;               // scalar f32
    float* partials = (float*)d_ws;                // 512 block partials

    hausdorff_tile_kernel<<<NB * 64, NTHR, 0, stream>>>(logits, y_true, partials);
    final_reduce_kernel<<<1, 256, 0, stream>>>(partials, out);
}